// GCN_77163382440458
// MI455X (gfx1250) — compile-verified
//
#include <hip/hip_runtime.h>
#include <hip/hip_bf16.h>

typedef float v2f __attribute__((ext_vector_type(2)));
typedef float v8f __attribute__((ext_vector_type(8)));

#define HDIM 64

// ---------------------------------------------------------------------------
// Degree / coefficient precompute
// ---------------------------------------------------------------------------
__global__ void k_init_deg(float* deg, int N) {
    int i = blockIdx.x * blockDim.x + threadIdx.x;
    if (i < N) deg[i] = 1.0f;  // self-loop weight
}

__global__ void k_deg_scatter(const int* dst, const float* ew, float* deg, int E) {
    int e = blockIdx.x * blockDim.x + threadIdx.x;
    if (e < E) unsafeAtomicAdd(&deg[dst[e]], ew[e]);
}

__global__ void k_coef(const float* deg, float* dinv, float* cself, int N) {
    int i = blockIdx.x * blockDim.x + threadIdx.x;
    if (i < N) {
        float d = deg[i];
        float r = (d > 0.0f) ? rsqrtf(d) : 0.0f;
        dinv[i]  = r;
        cself[i] = r * r;
    }
}

__global__ void k_edge_coef(const int* src, const int* dst, const float* ew,
                            const float* dinv, float* ce, int E) {
    int e = blockIdx.x * blockDim.x + threadIdx.x;
    if (e < E) ce[e] = dinv[src[e]] * ew[e] * dinv[dst[e]];
}

// ---------------------------------------------------------------------------
// Layer 1: H=1 input, trivial outer product  Htmp = x * W1 ; Agg = b1 + cself*Htmp
// ---------------------------------------------------------------------------
__global__ void k_layer1(const float* __restrict__ x, const float* __restrict__ W1,
                         const float* __restrict__ b1, const float* __restrict__ cself,
                         float* __restrict__ Htmp, float* __restrict__ Agg, int N) {
    int idx = blockIdx.x * blockDim.x + threadIdx.x;   // N*16 threads
    if (idx >= N * 16) return;
    int n = idx >> 4, g = idx & 15;
    float xv = x[n];
    float cs = cself[n];
    float4 w = ((const float4*)W1)[g];
    float4 b = ((const float4*)b1)[g];
    float4 h = make_float4(xv * w.x, xv * w.y, xv * w.z, xv * w.w);
    size_t o = (size_t)n * HDIM + g * 4;
    *(float4*)(Htmp + o) = h;
    *(float4*)(Agg + o)  = make_float4(b.x + cs * h.x, b.y + cs * h.y,
                                       b.z + cs * h.z, b.w + cs * h.w);
}

// ---------------------------------------------------------------------------
// WMMA GEMM:  D[16x64] = act(A[16x64]) @ W[64x64] per wave, f32 matrix core.
// W is staged once to LDS in fragment order, then the full 64x64 B operand set
// (64 fragments = 128 VGPRs) is preloaded into registers per wave. Tile loop
// then only streams A from global (relu fused) and issues 64 WMMAs.
// __launch_bounds__(256,1) gives the allocator the full VGPR file (no spills).
// MODE 0: store Htmp=D and Agg = bias + cself*D   (GCN layer, self-loop init)
// MODE 1: store Agg = D + bias                    (MLP head hidden)
// ---------------------------------------------------------------------------
template <int MODE, int RELU>
__global__ __launch_bounds__(256, 1)
void k_gemm64(const float* __restrict__ Ain, const float* __restrict__ W,
              const float* __restrict__ bias, const float* __restrict__ cself,
              float* __restrict__ Htmp, float* __restrict__ Agg, int ntiles) {
    __shared__ float2 lds_w[2048];                    // 16 KB, fragment-ordered W
    const int tid  = threadIdx.x;
    const int lane = tid & 31;
    const int wv   = tid >> 5;                        // 8 waves / block

    // cooperative fragment-order staging of W (64x64)
    #pragma unroll
    for (int i = 0; i < 8; ++i) {
        int p = tid + i * 256;                        // 2048 pairs total
        int l = p & 31, c = p >> 5;                   // c = kk*4 + t
        int kk = c >> 2, t = c & 3;
        int k = kk * 4 + 2 * (l >> 4);
        int n = t * 16 + (l & 15);
        lds_w[p] = make_float2(W[k * HDIM + n], W[(k + 1) * HDIM + n]);
    }
    __syncthreads();

    // preload ALL B fragments into registers (64 x float2 = 128 VGPRs)
    v2f breg[64];
    #pragma unroll
    for (int c = 0; c < 64; ++c) {
        float2 b2 = lds_w[c * 32 + lane];
        breg[c] = (v2f){b2.x, b2.y};
    }

    const int colBase = lane & 15;
    const int khalf   = 2 * (lane >> 4);

    for (int tile = blockIdx.x * 8 + wv; tile < ntiles; tile += gridDim.x * 8) {
        const int rowBase = tile * 16;
        const float* arow = Ain + (size_t)(rowBase + (lane & 15)) * HDIM + khalf;

        v8f acc[4];
        #pragma unroll
        for (int t = 0; t < 4; ++t) acc[t] = (v8f){0,0,0,0,0,0,0,0};

        #pragma unroll
        for (int kk = 0; kk < 16; ++kk) {
            float2 a2 = *(const float2*)(arow + kk * 4);
            if (RELU) { a2.x = fmaxf(a2.x, 0.0f); a2.y = fmaxf(a2.y, 0.0f); }
            v2f av = {a2.x, a2.y};
            #pragma unroll
            for (int t = 0; t < 4; ++t) {
                acc[t] = __builtin_amdgcn_wmma_f32_16x16x4_f32(
                    false, av, false, breg[kk * 4 + t], (short)0, acc[t], false, false);
            }
        }

        // epilogue: C/D layout -> row = rowBase + v + 8*(lane>=16), col = t*16 + lane%16
        float bv4[4];
        #pragma unroll
        for (int t = 0; t < 4; ++t) bv4[t] = bias[t * 16 + colBase];
        const int rowHalf = rowBase + 8 * (lane >> 4);
        #pragma unroll
        for (int v = 0; v < 8; ++v) {
            int row = rowHalf + v;
            float cs = (MODE == 0) ? cself[row] : 0.0f;
            #pragma unroll
            for (int t = 0; t < 4; ++t) {
                float D = acc[t][v];
                size_t o = (size_t)row * HDIM + t * 16 + colBase;
                if (MODE == 0) {
                    Htmp[o] = D;
                    Agg[o]  = bv4[t] + cs * D;
                } else {
                    Agg[o]  = D + bv4[t];
                }
            }
        }
    }
}

// ---------------------------------------------------------------------------
// Edge scatter: wave per edge, lane j handles columns 2j, 2j+1.
// Agg[dst] += ce * Htmp[src]   via hardware global_atomic_add_f32.
// Edge scalars made wave-uniform (readfirstlane) -> scalar loads.
// ---------------------------------------------------------------------------
__global__ void k_scatter(const float* __restrict__ Htmp, const float* __restrict__ ce,
                          const int* __restrict__ src, const int* __restrict__ dst,
                          float* __restrict__ Agg, int E) {
    int gtid = blockIdx.x * blockDim.x + threadIdx.x;
    int wave = gtid >> 5;
    int lane = threadIdx.x & 31;
    int nwaves = (gridDim.x * blockDim.x) >> 5;
    for (int e0 = wave; e0 < E; e0 += nwaves) {
        int e = __builtin_amdgcn_readfirstlane(e0);   // wave-uniform edge index
        int s = src[e], d = dst[e];
        float c = ce[e];
        float2 h = *(const float2*)(Htmp + (size_t)s * HDIM + 2 * lane);
        float* out = Agg + (size_t)d * HDIM + 2 * lane;
        unsafeAtomicAdd(out,     c * h.x);
        unsafeAtomicAdd(out + 1, c * h.y);
    }
}

// ---------------------------------------------------------------------------
// Final matvec: out[n] = fb2 + sum_j relu(T[n][j]) * fW2[j]
// ---------------------------------------------------------------------------
__global__ void k_head2(const float* __restrict__ T, const float* __restrict__ fW2,
                        const float* __restrict__ fb2, float* __restrict__ out, int N) {
    int n = blockIdx.x * blockDim.x + threadIdx.x;
    if (n >= N) return;
    const float4* tp = (const float4*)(T + (size_t)n * HDIM);
    const float4* wp = (const float4*)fW2;
    float acc = fb2[0];
    #pragma unroll
    for (int i = 0; i < 16; ++i) {
        float4 t = tp[i], w = wp[i];
        acc += fmaxf(t.x, 0.0f) * w.x + fmaxf(t.y, 0.0f) * w.y +
               fmaxf(t.z, 0.0f) * w.z + fmaxf(t.w, 0.0f) * w.w;
    }
    out[n] = acc;
}

// ---------------------------------------------------------------------------
// Host-side orchestration
// ---------------------------------------------------------------------------
extern "C" void kernel_launch(void* const* d_in, const int* in_sizes, int n_in,
                              void* d_out, int out_size, void* d_ws, size_t ws_size,
                              hipStream_t stream) {
    const float* x   = (const float*)d_in[0];
    const int*   ei  = (const int*)d_in[1];
    const float* ew  = (const float*)d_in[2];
    const float* Wl[5] = {(const float*)d_in[3], (const float*)d_in[5], (const float*)d_in[7],
                          (const float*)d_in[9], (const float*)d_in[11]};
    const float* bl[5] = {(const float*)d_in[4], (const float*)d_in[6], (const float*)d_in[8],
                          (const float*)d_in[10], (const float*)d_in[12]};
    const float* fW1 = (const float*)d_in[13];
    const float* fb1 = (const float*)d_in[14];
    const float* fW2 = (const float*)d_in[15];
    const float* fb2 = (const float*)d_in[16];
    float* out = (float*)d_out;

    const int N = in_sizes[0];        // 100000
    const int E = in_sizes[2];        // 1250000
    const int* esrc = ei;
    const int* edst = ei + E;

    // workspace carve-out
    char* ws = (char*)d_ws;
    float* Htmp = (float*)ws;                ws += (size_t)N * HDIM * sizeof(float);
    float* bufA = (float*)ws;                ws += (size_t)N * HDIM * sizeof(float);
    float* bufB = (float*)ws;                ws += (size_t)N * HDIM * sizeof(float);
    float* deg   = (float*)ws;               ws += (size_t)N * sizeof(float);
    float* dinv  = (float*)ws;               ws += (size_t)N * sizeof(float);
    float* cself = (float*)ws;               ws += (size_t)N * sizeof(float);
    float* ce    = (float*)ws;               ws += (size_t)E * sizeof(float);

    const int TB = 256;
    const int ntiles = N / 16;                          // 6250 (N % 16 == 0)
    const int gemmBlocks = (ntiles + 7) / 8;
    const int scatBlocks = (E + 7) / 8;                 // wave per edge

    // 1) normalization coefficients (shared across all layers)
    k_init_deg<<<(N + TB - 1) / TB, TB, 0, stream>>>(deg, N);
    k_deg_scatter<<<(E + TB - 1) / TB, TB, 0, stream>>>(edst, ew, deg, E);
    k_coef<<<(N + TB - 1) / TB, TB, 0, stream>>>(deg, dinv, cself, N);
    k_edge_coef<<<(E + TB - 1) / TB, TB, 0, stream>>>(esrc, edst, ew, dinv, ce, E);

    // 2) layer 1 (fan-in 1)
    k_layer1<<<(N * 16 + TB - 1) / TB, TB, 0, stream>>>(x, Wl[0], bl[0], cself, Htmp, bufA, N);
    k_scatter<<<scatBlocks, TB, 0, stream>>>(Htmp, ce, esrc, edst, bufA, E);

    // 3) layers 2..5: WMMA gemm (relu fused on load) + scatter, ping-pong Agg buffers
    float* aggIn = bufA;
    float* aggOut = bufB;
    for (int l = 1; l < 5; ++l) {
        k_gemm64<0, 1><<<gemmBlocks, TB, 0, stream>>>(aggIn, Wl[l], bl[l], cself,
                                                      Htmp, aggOut, ntiles);
        k_scatter<<<scatBlocks, TB, 0, stream>>>(Htmp, ce, esrc, edst, aggOut, E);
        float* t = aggIn; aggIn = aggOut; aggOut = t;
    }
    // aggIn now holds raw Agg5 (relu pending)

    // 4) MLP head: T = relu(Agg5) @ fW1 + fb1 (reuse Htmp as T), then matvec with fW2
    k_gemm64<1, 1><<<gemmBlocks, TB, 0, stream>>>(aggIn, fW1, fb1, nullptr,
                                                  nullptr, Htmp, ntiles);
    k_head2<<<(N + TB - 1) / TB, TB, 0, stream>>>(Htmp, fW2, fb2, out, N);
}